// Router_3530463117616
// MI455X (gfx1250) — compile-verified
//
#include <hip/hip_runtime.h>
#include <hip/hip_bf16.h>

typedef __attribute__((ext_vector_type(2))) float v2f;
typedef __attribute__((ext_vector_type(8))) float v8f;

#define DIM 4096
#define TOKENS 8192
#define NUM_EXPERTS 8
#define TOP_K 2
#define SLOTS (TOKENS * TOP_K)   // 16384

// ---------------------------------------------------------------------------
// Kernel 1: router scores via V_WMMA_F32_16X16X4_F32, then per-token top-2 +
// softmax over the two selected logits. One wave32 per 16-token tile.
//
// B trick: C columns 8..15 are never read (only lanes with row<8 spill scores
// to LDS), so lanes with row>=8 load W[row&7] unconditionally -> uniform
// control flow, no per-iteration EXEC save/restore around the B load.
// Four independent accumulators break the WMMA D->C dependency chain.
// ---------------------------------------------------------------------------
__global__ __launch_bounds__(32) void router_scores_topk(
    const float* __restrict__ x,   // [TOKENS, DIM]
    const float* __restrict__ W,   // [NUM_EXPERTS, DIM]
    int*   __restrict__ eids,      // [SLOTS] expert id per (token, k)
    float* __restrict__ probs)     // [SLOTS] softmax prob per (token, k)
{
  __shared__ float lds_scores[16][NUM_EXPERTS];

  const int lane  = threadIdx.x;          // 0..31
  const int tile  = blockIdx.x;           // 16 tokens per tile
  const int row   = lane & 15;            // M (token-in-tile) for A, N (expert) for B
  const int khalf = (lane >> 4) << 1;     // K sub-offset: 0 (lanes 0-15) / 2 (lanes 16-31)

  const float* xrow = x + (size_t)(tile * 16 + row) * DIM + khalf;
  const float* wrow = W + (size_t)(row & (NUM_EXPERTS - 1)) * DIM + khalf;

  v8f acc0 = {}, acc1 = {}, acc2 = {}, acc3 = {};

#pragma unroll 4
  for (int kb = 0; kb < DIM; kb += 16) {
    v2f a0 = *(const v2f*)(xrow + kb);
    v2f b0 = *(const v2f*)(wrow + kb);
    v2f a1 = *(const v2f*)(xrow + kb + 4);
    v2f b1 = *(const v2f*)(wrow + kb + 4);
    v2f a2 = *(const v2f*)(xrow + kb + 8);
    v2f b2 = *(const v2f*)(wrow + kb + 8);
    v2f a3 = *(const v2f*)(xrow + kb + 12);
    v2f b3 = *(const v2f*)(wrow + kb + 12);
    acc0 = __builtin_amdgcn_wmma_f32_16x16x4_f32(false, a0, false, b0,
                                                 (short)0, acc0, false, false);
    acc1 = __builtin_amdgcn_wmma_f32_16x16x4_f32(false, a1, false, b1,
                                                 (short)0, acc1, false, false);
    acc2 = __builtin_amdgcn_wmma_f32_16x16x4_f32(false, a2, false, b2,
                                                 (short)0, acc2, false, false);
    acc3 = __builtin_amdgcn_wmma_f32_16x16x4_f32(false, a3, false, b3,
                                                 (short)0, acc3, false, false);
  }
  acc0 += acc1;
  acc2 += acc3;
  acc0 += acc2;

  // C layout: VGPR r -> token M = r + 8*(lane>=16), column N = lane%16.
  if (row < NUM_EXPERTS) {
    const int mbase = (lane >> 4) << 3;
#pragma unroll
    for (int r = 0; r < 8; ++r)
      lds_scores[mbase + r][row] = acc0[r];
  }
  __syncthreads();

  if (lane < 16) {
    const int t = tile * 16 + lane;
    float best = -__builtin_inff(); int bi = 0;
    float sec  = -__builtin_inff(); int si = 0;
#pragma unroll
    for (int e = 0; e < NUM_EXPERTS; ++e) {
      const float s = lds_scores[lane][e];
      if (s > best)      { sec = best; si = bi; best = s; bi = e; }
      else if (s > sec)  { sec = s; si = e; }
    }
    // softmax over {best, sec}: exp(best-best)=1, exp(sec-best)
    const float e1 = __expf(sec - best);
    const float p0 = 1.0f / (1.0f + e1);
    eids[t * 2 + 0]  = bi;
    eids[t * 2 + 1]  = si;
    probs[t * 2 + 0] = p0;
    probs[t * 2 + 1] = 1.0f - p0;
  }
}

// ---------------------------------------------------------------------------
// Kernel 2: histogram over experts + exclusive prefix (single block).
// ---------------------------------------------------------------------------
__global__ __launch_bounds__(256) void count_and_offsets(
    const int* __restrict__ eids,
    int*   __restrict__ offsets,      // [NUM_EXPERTS] exclusive prefix
    float* __restrict__ out_counts)   // [NUM_EXPERTS] -> d_out (as float)
{
  __shared__ int cnt[NUM_EXPERTS];
  if (threadIdx.x < NUM_EXPERTS) cnt[threadIdx.x] = 0;
  __syncthreads();
  for (int i = threadIdx.x; i < SLOTS; i += blockDim.x)
    atomicAdd(&cnt[eids[i]], 1);
  __syncthreads();
  if (threadIdx.x == 0) {
    int run = 0;
    for (int e = 0; e < NUM_EXPERTS; ++e) {
      offsets[e]    = run;
      out_counts[e] = (float)cnt[e];
      run += cnt[e];
    }
  }
}

// ---------------------------------------------------------------------------
// Kernel 3: stable counting-sort ranks. One wave32 per expert; ordered chunk
// scan with wave32 ballot => stable by construction (== argsort stable=True).
// ---------------------------------------------------------------------------
__global__ __launch_bounds__(32) void stable_scatter(
    const int*   __restrict__ eids,
    const float* __restrict__ probs,
    const int*   __restrict__ offsets,
    int*   __restrict__ dst_tok,      // [SLOTS] source token per dest slot (scratch)
    float* __restrict__ out_scatter,  // [SLOTS] scatter_indices (as float) -> d_out
    float* __restrict__ out_scores)   // [SLOTS] scores_sorted -> d_out
{
  const int e    = blockIdx.x;
  const int lane = threadIdx.x;
  int run = offsets[e];
  for (int base = 0; base < SLOTS; base += 32) {
    const int  slot  = base + lane;
    const bool match = (eids[slot] == e);
    const unsigned mask = __builtin_amdgcn_ballot_w32(match);
    if (match) {
      const int d = run + __popc(mask & ((1u << lane) - 1u));
      const int tok = slot >> 1;
      dst_tok[d]     = tok;
      out_scatter[d] = (float)tok;
      out_scores[d]  = probs[slot];
    }
    run += __popc(mask);
  }
}

// ---------------------------------------------------------------------------
// Kernel 4: gather the selected token rows. Fully coalesced b128 copies; the
// 128 MB x tensor fits in the 192 MB L2, so these reads are mostly L2 hits.
// ---------------------------------------------------------------------------
__global__ __launch_bounds__(256) void gather_rows(
    const float* __restrict__ x,
    const int*   __restrict__ dst_tok,
    float* __restrict__ out_x)        // [SLOTS, DIM]
{
  const int d   = blockIdx.x;
  const int src = dst_tok[d];
  const float4* s = (const float4*)(x + (size_t)src * DIM);
  float4*       o = (float4*)(out_x + (size_t)d * DIM);
#pragma unroll 4
  for (int i = threadIdx.x; i < DIM / 4; i += 256)
    o[i] = s[i];
}

// ---------------------------------------------------------------------------
extern "C" void kernel_launch(void* const* d_in, const int* in_sizes, int n_in,
                              void* d_out, int out_size, void* d_ws, size_t ws_size,
                              hipStream_t stream) {
  const float* x = (const float*)d_in[0];   // [TOKENS, DIM]
  const float* W = (const float*)d_in[1];   // [NUM_EXPERTS, DIM]

  // d_out layout (reference return order, flat fp32):
  float* out        = (float*)d_out;
  float* out_x      = out;                               // SLOTS*DIM
  float* out_counts = out_x + (size_t)SLOTS * DIM;       // NUM_EXPERTS
  float* out_scat   = out_counts + NUM_EXPERTS;          // SLOTS
  float* out_scores = out_scat + SLOTS;                  // SLOTS

  // scratch layout
  int*   eids    = (int*)d_ws;                           // SLOTS
  float* probs   = (float*)(eids + SLOTS);               // SLOTS
  int*   offsets = (int*)(probs + SLOTS);                // NUM_EXPERTS
  int*   dst_tok = offsets + NUM_EXPERTS;                // SLOTS

  router_scores_topk<<<TOKENS / 16, 32, 0, stream>>>(x, W, eids, probs);
  count_and_offsets<<<1, 256, 0, stream>>>(eids, offsets, out_counts);
  stable_scatter<<<NUM_EXPERTS, 32, 0, stream>>>(eids, probs, offsets,
                                                 dst_tok, out_scat, out_scores);
  gather_rows<<<SLOTS, 256, 0, stream>>>(x, dst_tok, out_x);
}